// FeatureExtractor_1829656068304
// MI455X (gfx1250) — compile-verified
//
#include <hip/hip_runtime.h>
#include <math.h>

#define N_NODES 100000
#define N_EDGES 600000
#define DIM     128
#define NLAYERS 3
#define NGRAPH  64

typedef __attribute__((ext_vector_type(2))) float v2f;
typedef __attribute__((ext_vector_type(8))) float v8f;

// D = A(16x4 f32) * B(4x16 f32) + C(16x16 f32), full f32 precision matrix op.
__device__ __forceinline__ v8f wmma_f32(v2f a, v2f b, v8f c) {
  return __builtin_amdgcn_wmma_f32_16x16x4_f32(false, a, false, b, (short)0, c,
                                               false, false);
}

// CDNA5 async global->LDS copy (ASYNCcnt-tracked), GV addressing mode.
// lds_addr: wave-relative LDS byte address (low 32 bits of flat shared ptr).
__device__ __forceinline__ void async_load_b128(unsigned lds_addr,
                                                const void* gaddr) {
  asm volatile("global_load_async_to_lds_b128 %0, %1, off"
               :
               : "v"(lds_addr), "v"((unsigned long long)(uintptr_t)gaddr)
               : "memory");
}
__device__ __forceinline__ void wait_asynccnt0() {
  asm volatile("s_wait_asynccnt 0x0" ::: "memory");
}

// ---------------------------------------------------------------------------
// vn init: broadcast vn_table row 0 to all G rows
__global__ void vn_init_kernel(const float* __restrict__ vn_tab,
                               float* __restrict__ vn) {
  int t = blockIdx.x * blockDim.x + threadIdx.x;
  if (t < NGRAPH * DIM) vn[t] = vn_tab[t & (DIM - 1)];
}

// counts[g] = number of nodes in graph g
__global__ void count_kernel(const int* __restrict__ batch,
                             float* __restrict__ counts) {
  int i = blockIdx.x * blockDim.x + threadIdx.x;
  if (i < N_NODES) atomicAdd(counts + batch[i], 1.0f);
}

// agg[dst] += h[src]  (one thread = one edge * 4 features)
__global__ void scatter_kernel(const float* __restrict__ h,
                               const int* __restrict__ src,
                               const int* __restrict__ dst,
                               float* __restrict__ agg) {
  int t = blockIdx.x * blockDim.x + threadIdx.x;
  int e = t >> 5;
  if (e >= N_EDGES) return;
  int f = (t & 31) * 4;
  int s = src[e], d = dst[e];
  float4 v = *(const float4*)(h + (size_t)s * DIM + f);
  float* p = agg + (size_t)d * DIM + f;
  atomicAdd(p + 0, v.x);
  atomicAdd(p + 1, v.y);
  atomicAdd(p + 2, v.z);
  atomicAdd(p + 3, v.w);
}

// ---------------------------------------------------------------------------
// Fused GIN MLP over 16-row tiles:
//   T = (1+eps)*h + agg ; U = relu(T @ W1 + b1) ; hout = U @ W2 + b2
// Each of 8 waves owns a 16-column output tile; its W1/W2 B-fragments are
// loop-invariant and live in VGPRs (64+64 regs). h/agg tiles are staged to
// LDS with CDNA5 async-to-LDS loads; the [N,128] intermediate U never leaves
// LDS. Only ~25KB LDS -> multiple workgroups per WGP.
__global__ __launch_bounds__(256) void gin_mlp_kernel(
    const float* __restrict__ h, const float* __restrict__ agg,
    const float* __restrict__ eps_arr, int layer, const float* __restrict__ W1,
    const float* __restrict__ b1, const float* __restrict__ W2,
    const float* __restrict__ b2, float* __restrict__ hout) {
  __shared__ float lds_h[16 * DIM];  // 8 KB: h tile, then T in place
  __shared__ float lds_a[16 * DIM];  // 8 KB: agg tile
  __shared__ float lds_u[16 * DIM];  // 8 KB: U = relu(T@W1+b1)
  __shared__ float lds_b1[DIM];
  __shared__ float lds_b2[DIM];

  const int tid  = threadIdx.x;
  const int lane = tid & 31;
  const int wave = tid >> 5;
  const int half = lane >> 4;  // 0: K pair {0,1}, 1: K pair {2,3}
  const int l15  = lane & 15;
  const int colw = wave * 16;  // this wave's output column tile

  if (tid < DIM) {
    lds_b1[tid] = b1[tid];
    lds_b2[tid] = b2[tid];
  }
  const float onePlusEps = 1.0f + eps_arr[layer];

  // Preload this wave's B-fragments of W1/W2 into registers (loop-invariant).
  v2f bw1[32], bw2[32];
#pragma unroll
  for (int k = 0; k < 32; ++k) {
    const int kk = 4 * k + 2 * half;
    bw1[k].x = W1[(size_t)kk * DIM + colw + l15];
    bw1[k].y = W1[(size_t)(kk + 1) * DIM + colw + l15];
    bw2[k].x = W2[(size_t)kk * DIM + colw + l15];
    bw2[k].y = W2[(size_t)(kk + 1) * DIM + colw + l15];
  }
  __syncthreads();

  const unsigned lds_h_base = (unsigned)(uintptr_t)lds_h;
  const unsigned lds_a_base = (unsigned)(uintptr_t)lds_a;

  const int numTiles = N_NODES / 16;  // 6250, exact
  for (int tile = blockIdx.x; tile < numTiles; tile += gridDim.x) {
    const int row0 = tile * 16;
    const float* gh = h + (size_t)row0 * DIM;
    const float* ga = agg + (size_t)row0 * DIM;

    // Async-stage h and agg tiles (2048 floats each): 2 x b128 per thread.
#pragma unroll
    for (int i = 0; i < 2; ++i) {
      const int e4 = i * 256 + tid;  // float4 index
      async_load_b128(lds_h_base + e4 * 16, gh + e4 * 4);
      async_load_b128(lds_a_base + e4 * 16, ga + e4 * 4);
    }
    wait_asynccnt0();
    __syncthreads();

    // Fuse T = (1+eps)*h + agg in place in lds_h.
#pragma unroll
    for (int i = 0; i < 2; ++i) {
      const int e4 = i * 256 + tid;
      float4 hv = ((const float4*)lds_h)[e4];
      float4 av = ((const float4*)lds_a)[e4];
      hv.x = onePlusEps * hv.x + av.x;
      hv.y = onePlusEps * hv.y + av.y;
      hv.z = onePlusEps * hv.z + av.z;
      hv.w = onePlusEps * hv.w + av.w;
      ((float4*)lds_h)[e4] = hv;
    }
    __syncthreads();

    // GEMM1: U = relu(T @ W1 + b1)
    v8f acc;
    {
      const float bias = lds_b1[colw + l15];
#pragma unroll
      for (int r = 0; r < 8; ++r) acc[r] = bias;
    }
#pragma unroll
    for (int k = 0; k < 32; ++k) {
      const int kk = 4 * k + 2 * half;
      v2f a;
      a.x = lds_h[l15 * DIM + kk];
      a.y = lds_h[l15 * DIM + kk + 1];
      acc = wmma_f32(a, bw1[k], acc);
    }
#pragma unroll
    for (int r = 0; r < 8; ++r)
      lds_u[(r + 8 * half) * DIM + colw + l15] = fmaxf(acc[r], 0.0f);
    __syncthreads();

    // GEMM2: hout = U @ W2 + b2
    {
      const float bias = lds_b2[colw + l15];
#pragma unroll
      for (int r = 0; r < 8; ++r) acc[r] = bias;
    }
#pragma unroll
    for (int k = 0; k < 32; ++k) {
      const int kk = 4 * k + 2 * half;
      v2f a;
      a.x = lds_u[l15 * DIM + kk];
      a.y = lds_u[l15 * DIM + kk + 1];
      acc = wmma_f32(a, bw2[k], acc);
    }
#pragma unroll
    for (int r = 0; r < 8; ++r)
      hout[(size_t)(row0 + r + 8 * half) * DIM + colw + l15] = acc[r];
    __syncthreads();
  }
}

// ---------------------------------------------------------------------------
// vn_tmp[batch[i]] += relu(hout[i])
__global__ void pool_kernel(const float* __restrict__ hout,
                            const int* __restrict__ batch,
                            float* __restrict__ vn_tmp) {
  int t = blockIdx.x * blockDim.x + threadIdx.x;
  int i = t >> 5;
  if (i >= N_NODES) return;
  int f = (t & 31) * 4;
  int b = batch[i];
  float4 v = *(const float4*)(hout + (size_t)i * DIM + f);
  float* p = vn_tmp + (size_t)b * DIM + f;
  atomicAdd(p + 0, fmaxf(v.x, 0.0f));
  atomicAdd(p + 1, fmaxf(v.y, 0.0f));
  atomicAdd(p + 2, fmaxf(v.z, 0.0f));
  atomicAdd(p + 3, fmaxf(v.w, 0.0f));
}

// x = log1p(|x|) * x / (|x| + 1e-8)
__global__ void renorm_kernel(float* __restrict__ vn_tmp) {
  int t = blockIdx.x * blockDim.x + threadIdx.x;
  if (t >= NGRAPH * DIM) return;
  float x = vn_tmp[t];
  float a = fabsf(x);
  vn_tmp[t] = log1pf(a) * (x / (a + 1e-8f));
}

// Generic small GEMM with WMMA: C[M,Nn] = act(A[M,K] @ B[K,Nn] + bias)
// One wave per 16x16 output tile, grid-stride over tiles.
__global__ __launch_bounds__(256) void wmma_gemm_small(
    const float* __restrict__ A, const float* __restrict__ B,
    const float* __restrict__ bias, float* __restrict__ C, int M, int Nn,
    int K, int do_relu) {
  const int lane = threadIdx.x & 31;
  const int half = lane >> 4;
  const int l15  = lane & 15;
  const int wid    = (blockIdx.x * blockDim.x + threadIdx.x) >> 5;
  const int nwaves = (gridDim.x * blockDim.x) >> 5;
  const int tm = M / 16, tn = Nn / 16;
  for (int t = wid; t < tm * tn; t += nwaves) {
    const int row0 = (t / tn) * 16;
    const int col0 = (t % tn) * 16;
    v8f acc;
    const float bv = bias[col0 + l15];
#pragma unroll
    for (int r = 0; r < 8; ++r) acc[r] = bv;
    for (int k = 0; k < K; k += 4) {
      const int kk = k + 2 * half;
      v2f a, b;
      a.x = A[(size_t)(row0 + l15) * K + kk];
      a.y = A[(size_t)(row0 + l15) * K + kk + 1];
      b.x = B[(size_t)kk * Nn + col0 + l15];
      b.y = B[(size_t)(kk + 1) * Nn + col0 + l15];
      acc = wmma_f32(a, b, acc);
    }
#pragma unroll
    for (int r = 0; r < 8; ++r) {
      float v = acc[r];
      if (do_relu) v = fmaxf(v, 0.0f);
      C[(size_t)(row0 + r + 8 * half) * Nn + col0 + l15] = v;
    }
  }
}

// h = relu(hout + vn[batch]);  z_acc += h
__global__ void combine_kernel(const float* __restrict__ hout,
                               const float* __restrict__ vn,
                               const int* __restrict__ batch,
                               float* __restrict__ h,
                               float* __restrict__ z_acc) {
  int t = blockIdx.x * blockDim.x + threadIdx.x;
  int i = t >> 5;
  if (i >= N_NODES) return;
  int f = (t & 31) * 4;
  int b = batch[i];
  size_t idx = (size_t)i * DIM + f;
  float4 hv = *(const float4*)(hout + idx);
  float4 vv = *(const float4*)(vn + (size_t)b * DIM + f);
  float4 r;
  r.x = fmaxf(hv.x + vv.x, 0.0f);
  r.y = fmaxf(hv.y + vv.y, 0.0f);
  r.z = fmaxf(hv.z + vv.z, 0.0f);
  r.w = fmaxf(hv.w + vv.w, 0.0f);
  *(float4*)(h + idx) = r;
  float4 z = *(const float4*)(z_acc + idx);
  z.x += r.x; z.y += r.y; z.z += r.z; z.w += r.w;
  *(float4*)(z_acc + idx) = z;
}

// z = z_acc / L (in place in d_out); gsum[batch[i]] += z
__global__ void finalize1_kernel(const int* __restrict__ batch,
                                 float* __restrict__ z,
                                 float* __restrict__ gsum) {
  int t = blockIdx.x * blockDim.x + threadIdx.x;
  int i = t >> 5;
  if (i >= N_NODES) return;
  int f = (t & 31) * 4;
  size_t idx = (size_t)i * DIM + f;
  const float inv = 1.0f / (float)NLAYERS;
  float4 zv = *(const float4*)(z + idx);
  zv.x *= inv; zv.y *= inv; zv.z *= inv; zv.w *= inv;
  *(float4*)(z + idx) = zv;
  int b = batch[i];
  float* p = gsum + (size_t)b * DIM + f;
  atomicAdd(p + 0, zv.x);
  atomicAdd(p + 1, zv.y);
  atomicAdd(p + 2, zv.z);
  atomicAdd(p + 3, zv.w);
}

__global__ void finalize2_kernel(const float* __restrict__ gsum,
                                 const float* __restrict__ counts,
                                 float* __restrict__ g_out) {
  int t = blockIdx.x * blockDim.x + threadIdx.x;
  if (t >= NGRAPH * DIM) return;
  g_out[t] = gsum[t] / fmaxf(counts[t >> 7], 1.0f);
}

// ---------------------------------------------------------------------------
extern "C" void kernel_launch(void* const* d_in, const int* in_sizes, int n_in,
                              void* d_out, int out_size, void* d_ws,
                              size_t ws_size, hipStream_t stream) {
  (void)in_sizes; (void)n_in; (void)ws_size;

  const float* x      = (const float*)d_in[0];
  const int*   eidx   = (const int*)d_in[1];
  const int*   batch  = (const int*)d_in[2];
  const float* vn_tab = (const float*)d_in[3];
  const float* eps    = (const float*)d_in[4];
  const float* W1     = (const float*)d_in[5];
  const float* b1     = (const float*)d_in[6];
  const float* W2     = (const float*)d_in[7];
  const float* b2     = (const float*)d_in[8];
  const float* Wv1    = (const float*)d_in[9];
  const float* bv1    = (const float*)d_in[10];
  const float* Wv2    = (const float*)d_in[11];
  const float* bv2    = (const float*)d_in[12];

  const int* src = eidx;             // edge_index[0]
  const int* dst = eidx + N_EDGES;   // edge_index[1]

  float* ws = (float*)d_ws;
  size_t off = 0;
  float* h      = ws + off; off += (size_t)N_NODES * DIM;
  float* agg    = ws + off; off += (size_t)N_NODES * DIM;
  float* hout   = ws + off; off += (size_t)N_NODES * DIM;
  float* vn     = ws + off; off += (size_t)NGRAPH * DIM;
  float* vn_tmp = ws + off; off += (size_t)NGRAPH * DIM;
  float* vn_h   = ws + off; off += (size_t)NGRAPH * 2 * DIM;
  float* gsum   = ws + off; off += (size_t)NGRAPH * DIM;
  float* counts = ws + off; off += (size_t)NGRAPH;

  float* z_out = (float*)d_out;
  float* g_out = z_out + (size_t)N_NODES * DIM;

  hipMemsetAsync(d_out, 0, (size_t)out_size * sizeof(float), stream);
  hipMemsetAsync(gsum, 0, ((size_t)NGRAPH * DIM + NGRAPH) * sizeof(float),
                 stream);  // gsum + counts are contiguous
  hipMemcpyAsync(h, x, (size_t)N_NODES * DIM * sizeof(float),
                 hipMemcpyDeviceToDevice, stream);
  vn_init_kernel<<<(NGRAPH * DIM) / 256, 256, 0, stream>>>(vn_tab, vn);
  count_kernel<<<(N_NODES + 255) / 256, 256, 0, stream>>>(batch, counts);

  for (int l = 0; l < NLAYERS; ++l) {
    hipMemsetAsync(agg, 0, (size_t)N_NODES * DIM * sizeof(float), stream);
    scatter_kernel<<<(N_EDGES * 32) / 256, 256, 0, stream>>>(h, src, dst, agg);
    gin_mlp_kernel<<<1024, 256, 0, stream>>>(
        h, agg, eps, l, W1 + (size_t)l * DIM * DIM, b1 + (size_t)l * DIM,
        W2 + (size_t)l * DIM * DIM, b2 + (size_t)l * DIM, hout);
    hipMemcpyAsync(vn_tmp, vn, (size_t)NGRAPH * DIM * sizeof(float),
                   hipMemcpyDeviceToDevice, stream);
    pool_kernel<<<(N_NODES * 32 + 255) / 256, 256, 0, stream>>>(hout, batch,
                                                                vn_tmp);
    renorm_kernel<<<(NGRAPH * DIM) / 256, 256, 0, stream>>>(vn_tmp);
    // vn_h = relu(vn_tmp @ Wv1 + bv1)   [64,128]x[128,256]
    wmma_gemm_small<<<8, 256, 0, stream>>>(
        vn_tmp, Wv1 + (size_t)l * DIM * 2 * DIM, bv1 + (size_t)l * 2 * DIM,
        vn_h, NGRAPH, 2 * DIM, DIM, 1);
    // vn = vn_h @ Wv2 + bv2             [64,256]x[256,128]
    wmma_gemm_small<<<8, 256, 0, stream>>>(
        vn_h, Wv2 + (size_t)l * 2 * DIM * DIM, bv2 + (size_t)l * DIM, vn,
        NGRAPH, DIM, 2 * DIM, 0);
    combine_kernel<<<(N_NODES * 32 + 255) / 256, 256, 0, stream>>>(
        hout, vn, batch, h, z_out);
  }

  finalize1_kernel<<<(N_NODES * 32 + 255) / 256, 256, 0, stream>>>(batch, z_out,
                                                                   gsum);
  finalize2_kernel<<<(NGRAPH * DIM) / 256, 256, 0, stream>>>(gsum, counts,
                                                             g_out);
}